// LogmelIntensity_Extractor_22067541967148
// MI455X (gfx1250) — compile-verified
//
#include <hip/hip_runtime.h>
#include <math.h>

typedef float v2f __attribute__((ext_vector_type(2)));
typedef float v8f __attribute__((ext_vector_type(8)));

#define NFFT   1024
#define FBINS  513            // NFFT/2 + 1
#define NMELS  64
#define HOP    300
#define TFRM   401            // 1 + 120000/300
#define TP     416            // frames padded to 13*32
#define JP     1040           // spectral row: 513 Re + 513 Im = 1026 -> pad to 1040
#define JPT    1280           // basis rows padded to 80 N-tiles (branch-free 2xN blocking)
#define XLEN   120000
#define NB     32
#define ASTRIDE 1028          // LDS frame row stride (floats): bank-conflict-free A fetch
#define PI_D   3.14159265358979323846

// d_ws float layout
#define WS_BT   0                           // Bt[JPT][NFFT]  (basis, transposed, window folded)
#define WS_GC   (WS_BT + JPT*NFFT)          // Gc[FBINS][NMELS]
#define WS_GS   (WS_GC + FBINS*NMELS)       // Gs[FBINS][NMELS]
#define WS_SPEC (WS_GS + FBINS*NMELS)       // spec[NB*2][TP][JP]

// ---------------- table builders ----------------

// Bt[j][n]: j<513 -> window[n]*cos(2pi j n/1024); 513<=j<1026 -> -window[n]*sin(...); pad rows = 0
__global__ void build_basis(const float* __restrict__ window, float* __restrict__ Bt) {
  int i = blockIdx.x * 256 + threadIdx.x;
  if (i >= JPT * NFFT) return;
  int j = i >> 10;
  int n = i & (NFFT - 1);
  float v = 0.0f;
  if (j < 2 * FBINS) {
    int k = (j < FBINS) ? j : (j - FBINS);
    int p = (k * n) & (NFFT - 1);                 // exact phase reduction
    float th = (float)(2.0 * PI_D / NFFT) * (float)p;
    float s, c;
    sincosf(th, &s, &c);
    v = window[n] * ((j < FBINS) ? c : -s);
  }
  Bt[i] = v;
}

// 64-lag inverse-DFT tables: gcc[jj] = sum_k ur[k]*Gc[k][jj] + ui[k]*Gs[k][jj]
__global__ void build_gcc(float* __restrict__ Gc, float* __restrict__ Gs) {
  int i = blockIdx.x * 256 + threadIdx.x;
  if (i >= FBINS * NMELS) return;
  int k  = i / NMELS;
  int jj = i - k * NMELS;
  int nj = (jj < 32) ? (NFFT - 32 + jj) : (jj - 32);  // lag order: [-32..-1, 0..31]
  float ck = (k == 0 || k == FBINS - 1) ? 1.0f : 2.0f;
  int p = (k * nj) & (NFFT - 1);
  float th = (float)(2.0 * PI_D / NFFT) * (float)p;
  float s, c;
  sincosf(th, &s, &c);
  Gc[i] =  ck * c * (1.0f / NFFT);
  Gs[i] = -ck * s * (1.0f / NFFT);
}

// ---------------- STFT as WMMA f32 GEMM, 2x2 tile-blocked ----------------
// grid: (TP/32, 5, NB*2), block 256 (8 waves).
// C[m][j] = sum_n frame[m][n] * Bt[j][n].
// Each wave: 2 M-tiles (frames 0-15 / 16-31 of the staged 32) x 2 N-tiles
// (tile w and tile w+8 of the 16-tile group) -> 4 accumulators; per K-step
// only 2 A-loads + 2 B-loads feed 4 WMMAs (halves LDS + L2 traffic per FLOP).
__global__ void __launch_bounds__(256)
stft_wmma(const float* __restrict__ x, const float* __restrict__ Bt,
          float* __restrict__ spec) {
  __shared__ float ldsA[32 * ASTRIDE];   // ~128.5 KB, conflict-free row stride
  const int mgrp = blockIdx.x;                        // group of 32 frames
  const int bc   = blockIdx.z;                        // b*2 + ch
  const float* xc = x + (size_t)((bc >> 1) * 4 + (bc & 1)) * XLEN;
  const int tid = threadIdx.x;

  // stage 32 frames, reflect padding (pad = NFFT/2), coalesced in n
  for (int i = tid; i < 32 * NFFT; i += 256) {
    int mf = i >> 10;
    int n  = i & (NFFT - 1);
    int p  = (mgrp * 32 + mf) * HOP + n - (NFFT / 2);
    p = (p < 0) ? -p : p;
    p = (p >= XLEN) ? (2 * (XLEN - 1) - p) : p;
    ldsA[mf * ASTRIDE + n] = xc[p];
  }
  __syncthreads();

  const int wave = tid >> 5;
  const int lane = tid & 31;
  const int j0a = (blockIdx.y * 16 + wave) * 16;      // N-tile w of group
  const int j0b = j0a + 128;                          // N-tile w+8 of group
  if (j0a >= JP) return;                              // after the only barrier

  const int half = lane >> 4;                         // 0/1 -> K pair select
  const int l15  = lane & 15;
  // A: M=l15 (frame), K = k0 + 2*half + {0,1}   (ISA 16x4 f32 A layout)
  const float* arow0 = &ldsA[l15 * ASTRIDE + 2 * half];         // frames 0..15
  const float* arow1 = arow0 + 16 * ASTRIDE;                    // frames 16..31
  // B: N=l15 (bin),   K = k0 + 2*half + {0,1}   (rows striped across lane halves)
  const float* brow0 = &Bt[(size_t)(j0a + l15) * NFFT + 2 * half];
  const float* brow1 = &Bt[(size_t)(j0b + l15) * NFFT + 2 * half];

  v8f acc00 = {}, acc10 = {}, acc01 = {}, acc11 = {};
#pragma unroll 4
  for (int k0 = 0; k0 < NFFT; k0 += 4) {
    v2f b0 = *(const v2f*)(brow0 + k0);
    v2f b1 = *(const v2f*)(brow1 + k0);
    v2f a0 = *(const v2f*)(arow0 + k0);
    v2f a1 = *(const v2f*)(arow1 + k0);
    acc00 = __builtin_amdgcn_wmma_f32_16x16x4_f32(false, a0, false, b0,
                                                  (short)0, acc00, false, false);
    acc10 = __builtin_amdgcn_wmma_f32_16x16x4_f32(false, a1, false, b0,
                                                  (short)0, acc10, false, false);
    acc01 = __builtin_amdgcn_wmma_f32_16x16x4_f32(false, a0, false, b1,
                                                  (short)0, acc01, false, false);
    acc11 = __builtin_amdgcn_wmma_f32_16x16x4_f32(false, a1, false, b1,
                                                  (short)0, acc11, false, false);
  }

  // D layout: VGPR r -> row r + 8*half, col l15
  const size_t rowbase = ((size_t)bc * TP + (size_t)mgrp * 32) * JP;
#pragma unroll
  for (int r = 0; r < 8; ++r) {
    int mloc = r + 8 * half;
    spec[rowbase + (size_t)mloc * JP        + j0a + l15] = acc00[r];
    spec[rowbase + (size_t)(mloc + 16) * JP + j0a + l15] = acc10[r];
  }
  if (j0b < JP) {
#pragma unroll
    for (int r = 0; r < 8; ++r) {
      int mloc = r + 8 * half;
      spec[rowbase + (size_t)mloc * JP        + j0b + l15] = acc01[r];
      spec[rowbase + (size_t)(mloc + 16) * JP + j0b + l15] = acc11[r];
    }
  }
}

// ---------------- per-frame features ----------------
// grid (TFRM, NB), block 256
__global__ void __launch_bounds__(256)
post(const float* __restrict__ spec, const float* __restrict__ melW,
     const float* __restrict__ Gc, const float* __restrict__ Gs,
     float* __restrict__ out) {
  __shared__ float sx[2 * FBINS];     // ch0: Re[0..512], Im[513..1025]
  __shared__ float sr[2 * FBINS];     // ch1
  __shared__ float ur[FBINS], ui[FBINS];   // PHAT unit vectors
  const int t = blockIdx.x;
  const int b = blockIdx.y;
  const int tid = threadIdx.x;

  const float* px = spec + ((size_t)(b * 2 + 0) * TP + t) * JP;
  const float* pr = spec + ((size_t)(b * 2 + 1) * TP + t) * JP;
  for (int i = tid; i < 2 * FBINS; i += 256) { sx[i] = px[i]; sr[i] = pr[i]; }
  __syncthreads();

  for (int k = tid; k < FBINS; k += 256) {
    float xr = sx[k], xi = sx[FBINS + k];
    float rr = sr[k], ri = sr[FBINS + k];
    float a  = xr * rr + xi * ri;       // Re(Px * conj(Pr))
    float bb = xi * rr - xr * ri;       // Im(Px * conj(Pr))
    float m2 = a * a + bb * bb;
    if (m2 > 0.0f) {
      float inv = rsqrtf(m2);
      ur[k] = a * inv; ui[k] = bb * inv;
    } else {                            // angle(0) = 0 -> unit = (1, 0)
      ur[k] = 1.0f; ui[k] = 0.0f;
    }
  }
  __syncthreads();

  const size_t fs = (size_t)TFRM * NMELS;
  if (tid < NMELS) {
    int m = tid;
    float ar = 0.f, ai = 0.f, br = 0.f, bi = 0.f;
    for (int k = 0; k < FBINS; ++k) {
      float w = melW[k * NMELS + m];
      ar += sx[k] * w;         ai += sx[FBINS + k] * w;
      br += sr[k] * w;         bi += sr[FBINS + k] * w;
    }
    float mag  = sqrtf(ar * ar + ai * ai);
    float rmag = sqrtf(br * br + bi * bi);
    float ild  = mag / (rmag + 1e-20f);
    float dphi = atan2f(ai, ar) - atan2f(bi, br);
    float sp, cp; sincosf(dphi, &sp, &cp);
    float lg  = 10.0f * log10f(fmaxf(mag,  1e-10f));
    float lgr = 10.0f * log10f(fmaxf(rmag, 1e-10f));
    size_t ob = ((size_t)b * 6 * TFRM + t) * NMELS + m;   // f = 0 slot
    out[ob]          = lg;
    out[ob + fs]     = lgr;
    out[ob + 2 * fs] = ild;
    out[ob + 3 * fs] = sp;
    out[ob + 4 * fs] = cp;
  } else if (tid < 2 * NMELS) {
    int jj = tid - NMELS;
    float g = 0.f;
    for (int k = 0; k < FBINS; ++k)
      g += ur[k] * Gc[k * NMELS + jj] + ui[k] * Gs[k * NMELS + jj];
    out[((size_t)(b * 6 + 5) * TFRM + t) * NMELS + jj] = g;
  }
}

// ---------------- launch ----------------
extern "C" void kernel_launch(void* const* d_in, const int* in_sizes, int n_in,
                              void* d_out, int out_size, void* d_ws, size_t ws_size,
                              hipStream_t stream) {
  const float* x      = (const float*)d_in[0];   // (32, 4, 120000)
  const float* window = (const float*)d_in[1];   // (1024,)
  const float* melW   = (const float*)d_in[2];   // (513, 64)
  float* ws   = (float*)d_ws;
  float* Bt   = ws + WS_BT;
  float* Gc   = ws + WS_GC;
  float* Gs   = ws + WS_GS;
  float* spec = ws + WS_SPEC;

  build_basis<<<(JPT * NFFT + 255) / 256, 256, 0, stream>>>(window, Bt);
  build_gcc<<<(FBINS * NMELS + 255) / 256, 256, 0, stream>>>(Gc, Gs);

  dim3 g2(TP / 32, 5, NB * 2);   // 13 M-groups x 5 N-groups(16 tiles) x 64 (b,ch)
  stft_wmma<<<g2, 256, 0, stream>>>(x, Bt, spec);

  dim3 g3(TFRM, NB);
  post<<<g3, 256, 0, stream>>>(spec, melW, Gc, Gs, (float*)d_out);
}